// AtomPooling_SA_17978733101773
// MI455X (gfx1250) — compile-verified
//
#include <hip/hip_runtime.h>

#define NSEG 2048
#define DIM 256
#define OUT_DIM 256
#define CH 8
#define NWAVES 8

typedef __attribute__((ext_vector_type(2))) float v2f;
typedef __attribute__((ext_vector_type(8))) float v8f;

// ---------------------------------------------------------------------------
// Kernel 1: fused per-segment attention pooling with single-pass online
// softmax. One block (8 wave32) per segment; thread t owns channel t.
// Reads atom_features exactly ONCE (256 MB -> ~11us at 23.3 TB/s).
// ---------------------------------------------------------------------------
__global__ __launch_bounds__(256) void seg_attn_pool_kernel(
    const float* __restrict__ feat,       // [N_ATOMS, DIM]
    const long long* __restrict__ offs,   // [NSEG+1] int64 offsets
    const float* __restrict__ Wa,         // [DIM]
    const float* __restrict__ ba,         // [1]
    float* __restrict__ pooled)           // [NSEG, DIM] (workspace)
{
    __shared__ float part[NWAVES * CH];

    const int seg  = blockIdx.x;
    const int t    = threadIdx.x;
    const int lane = t & 31;
    const int wid  = t >> 5;

    const long long start = offs[seg];
    const long long end   = offs[seg + 1];
    const float wa  = Wa[t];
    const float bav = ba[0];

    // online softmax state (identical across all threads of the block)
    float m   = -3.402823466e+38f;   // running max
    float l   = 0.0f;                // running denom
    float acc = 0.0f;                // running weighted feature sum (channel t)

    for (long long base = start; base < end; base += CH) {
        const long long rem = end - base;
        const int nv = rem < (long long)CH ? (int)rem : CH;

        // load chunk features (coalesced: thread t reads channel t) + logit partials
        float f[CH], p[CH];
        #pragma unroll
        for (int a = 0; a < CH; ++a) {
            f[a] = (a < nv) ? feat[(base + a) * (long long)DIM + t] : 0.0f;
            p[a] = f[a] * wa;
        }
        if (base + CH < end)
            __builtin_prefetch(&feat[(base + CH) * (long long)DIM + t], 0, 0);

        // wave32 tree reduction of logit partials
        #pragma unroll
        for (int a = 0; a < CH; ++a) {
            #pragma unroll
            for (int off = 16; off > 0; off >>= 1)
                p[a] += __shfl_xor(p[a], off, 32);
        }
        if (lane == 0) {
            #pragma unroll
            for (int a = 0; a < CH; ++a) part[wid * CH + a] = p[a];
        }
        __syncthreads();

        // cross-wave combine (broadcast reads, conflict-free) -> full logits
        float logit[CH];
        float cmax = -3.402823466e+38f;
        #pragma unroll
        for (int a = 0; a < CH; ++a) {
            float s = bav;
            #pragma unroll
            for (int w = 0; w < NWAVES; ++w) s += part[w * CH + a];
            logit[a] = s;
            if (a < nv) cmax = fmaxf(cmax, s);
        }
        __syncthreads();

        // online softmax update; chunk features are still in registers,
        // so no second read of atom_features is ever needed.
        const float newm  = fmaxf(m, cmax);
        const float scale = __expf(m - newm);   // first chunk: exp(-huge)=0
        l   *= scale;
        acc *= scale;
        #pragma unroll
        for (int a = 0; a < CH; ++a) {
            if (a < nv) {
                const float w = __expf(logit[a] - newm);
                l += w;
                acc = fmaf(w, f[a], acc);
            }
        }
        m = newm;
    }

    pooled[(long long)seg * DIM + t] = (l > 0.0f) ? (acc / l) : 0.0f;
}

// ---------------------------------------------------------------------------
// Kernel 2: out = pooled @ Wo + bo  (2048x256x256 f32 GEMM) using
// V_WMMA_F32_16X16X4_F32. One 16x16 output tile per wave, K stepped by 4.
// f32 WMMA keeps exact f32 accumulation (matches reference precision).
// A/B live in L2 (2 MB + 256 KB), so the repeated reads are cheap.
// ---------------------------------------------------------------------------
__global__ __launch_bounds__(256) void proj_wmma_kernel(
    const float* __restrict__ pooled,   // [NSEG, DIM]
    const float* __restrict__ Wo,       // [DIM, OUT_DIM]
    const float* __restrict__ bo,       // [OUT_DIM]
    float* __restrict__ out)            // [NSEG, OUT_DIM]
{
    const int lane = threadIdx.x & 31;
    const int wid  = threadIdx.x >> 5;
    const int tile = blockIdx.x * NWAVES + wid;     // 2048 tiles total
    const int m0 = (tile >> 4) * 16;                // 128 M-tiles
    const int n0 = (tile & 15) * 16;                // 16  N-tiles

    const int row   = lane & 15;          // A row / D column index
    const int col   = lane & 15;          // B/D column index
    const int khalf = (lane >> 4) * 2;    // upper half-wave holds K+2,K+3

    const float* __restrict__ arow = pooled + (long long)(m0 + row) * DIM;

    v8f c = {};
    for (int k0 = 0; k0 < DIM; k0 += 4) {
        // A 16x4 f32 frag: VGPR0 = K=0 (lanes 0-15) / K=2 (lanes 16-31),
        //                  VGPR1 = K=1 / K=3  -> contiguous pair per lane.
        const float2 av = *(const float2*)(arow + k0 + khalf);
        v2f a; a.x = av.x; a.y = av.y;
        // B 4x16 f32 frag: rows striped across lanes within a VGPR.
        v2f b;
        b.x = Wo[(long long)(k0 + khalf) * OUT_DIM + n0 + col];
        b.y = Wo[(long long)(k0 + khalf + 1) * OUT_DIM + n0 + col];
        // 8 args: (neg_a, A, neg_b, B, c_mod, C, reuse_a, reuse_b)
        c = __builtin_amdgcn_wmma_f32_16x16x4_f32(
                false, a, false, b, (short)0, c, false, false);
    }

    // D 16x16 f32 layout: VGPR r -> (M = r + 8*(lane>=16), N = lane%16)
    const float bv   = bo[n0 + col];
    const int mbase  = m0 + ((lane >> 4) << 3);
    #pragma unroll
    for (int r = 0; r < 8; ++r)
        out[(long long)(mbase + r) * OUT_DIM + n0 + col] = c[r] + bv;
}

// ---------------------------------------------------------------------------
extern "C" void kernel_launch(void* const* d_in, const int* in_sizes, int n_in,
                              void* d_out, int out_size, void* d_ws, size_t ws_size,
                              hipStream_t stream) {
    const float*     feat = (const float*)d_in[0];       // [262144,256] f32
    const long long* offs = (const long long*)d_in[1];   // [2049] int64
    const float*     Wa   = (const float*)d_in[2];       // [256,1]
    const float*     ba   = (const float*)d_in[3];       // [1]
    const float*     Wo   = (const float*)d_in[4];       // [256,256]
    const float*     bo   = (const float*)d_in[5];       // [256]
    float* out    = (float*)d_out;                       // [2048,256]
    float* pooled = (float*)d_ws;                        // 2 MB scratch

    seg_attn_pool_kernel<<<NSEG, 256, 0, stream>>>(feat, offs, Wa, ba, pooled);

    // 2048x256 outputs / (16x16 per wave) = 2048 tiles / 8 waves = 256 blocks
    proj_wmma_kernel<<<(NSEG / 16) * (OUT_DIM / 16) / NWAVES, 256, 0, stream>>>(
        pooled, Wo, bo, out);
}